// SimRel_17763984736731
// MI455X (gfx1250) — compile-verified
//
#include <hip/hip_runtime.h>
#include <math.h>
#include <stdint.h>

// SimRel forward for MI455X (gfx1250, wave32).
// B=16, T=1024, D=2048, K=64 -> N=16384 tokens.
// Chunked sequential scan: per 64-token chunk, two fp32 WMMA GEMMs
// (prototype-state x chunk, and chunk Gram matrix), then an O(64^2)
// sequential correction that reproduces the token-by-token semantics.
// Staging uses CDNA5 async global->LDS (ASYNCcnt) instead of load+ds_store.

#define BB 16
#define TT 1024
#define DD 2048
#define KK 64
#define NN (BB * TT)
#define CH 64            // tokens per chunk
#define NCH (NN / CH)    // 256 chunks
#define KS 32            // K-slice staged in LDS
#define EPSV 1e-8f

typedef __attribute__((ext_vector_type(2))) float v2f;
typedef __attribute__((ext_vector_type(8))) float v8f;

// CDNA5 async copy: global memory -> LDS, 16 bytes per lane, tracked by ASYNCcnt.
// vdst = LDS byte address (VGPR), vaddr = 64-bit global address (VGPR pair).
__device__ __forceinline__ void async_ld_b128(void* lds_ptr, const void* gptr) {
    uint32_t laddr = (uint32_t)(uintptr_t)lds_ptr;   // low 32 bits = LDS offset
    uint64_t gaddr = (uint64_t)(uintptr_t)gptr;
    asm volatile("global_load_async_to_lds_b128 %0, %1, off"
                 :: "v"(laddr), "v"(gaddr) : "memory");
}
__device__ __forceinline__ void wait_async0() {
    asm volatile("s_wait_asynccnt 0" ::: "memory");
}

__global__ __launch_bounds__(512)
void simrel_cdna5_kernel(const float* __restrict__ inputs,      // [N, D]
                         const float* __restrict__ class_avgs,  // [K, D]
                         const int*   __restrict__ labels,      // [N]
                         float* __restrict__ out,               // [N, K]
                         float* __restrict__ S)                 // ws: [K, D] prototype sums
{
    __shared__ float sS[64][KS];      // prototype K-slice
    __shared__ float sV[64][KS];      // token K-slice
    __shared__ float simsT[64][64];   // dot(S0_k, v_j)  [class][token]
    __shared__ float gram[64][64];    // dot(v_i, v_j)   [token][token]
    __shared__ float n2[64];          // ||S_k||^2 (persistent across chunks)
    __shared__ int   lab[CH];
    __shared__ int   wasSeen[64];     // seen before this chunk
    __shared__ int   firstIn[64];     // first in-chunk occurrence (64 = none yet)
    __shared__ int   seenP[64];       // persistent seen flags

    const int tid  = threadIdx.x;
    const int lane = tid & 31;
    const int wave = tid >> 5;        // 0..15
    const int mi   = wave >> 2;       // row tile (A rows)
    const int ni   = wave & 3;        // col tile (B cols)
    const int l16  = lane & 15;
    const int hi   = lane >> 4;       // 0: K pair {0,1}, 1: K pair {2,3}

    // ---- per-launch init: S = class_avgs; n2 = row norms; seen = 0
    {
        const float4* src = (const float4*)class_avgs;
        float4*       dst = (float4*)S;
        for (int i = tid; i < KK * DD / 4; i += (int)blockDim.x) dst[i] = src[i];
        if (tid < KK) {
            float a = 0.f;
            for (int d = 0; d < DD; ++d) {
                float x = class_avgs[tid * DD + d];
                a += x * x;
            }
            n2[tid]    = a;
            seenP[tid] = 0;
        }
    }
    __syncthreads();

    for (int c = 0; c < NCH; ++c) {
        const int base = c * CH;
        if (tid < KK) { wasSeen[tid] = seenP[tid]; firstIn[tid] = 64; }
        if (tid < CH) lab[tid] = labels[base + tid];
        __syncthreads();

        // ---- two GEMMs over K, fp32 WMMA 16x16x4
        v8f acc1 = {};   // S0 x V^T  (classes x tokens)
        v8f acc2 = {};   // V  x V^T  (tokens  x tokens)
        for (int ks = 0; ks < DD; ks += KS) {
            __syncthreads();   // previous slice fully consumed
            {
                // 512 threads: one async b128 per array each (64 rows x 32 cols)
                int row = tid >> 3;
                int col = (tid & 7) << 2;
                async_ld_b128(&sV[row][col],
                              &inputs[(size_t)(base + row) * DD + ks + col]);
                async_ld_b128(&sS[row][col],
                              &S[(size_t)row * DD + ks + col]);
                if (c + 1 < NCH)  // warm next chunk into cache
                    __builtin_prefetch(&inputs[(size_t)(base + CH + row) * DD + ks + col], 0, 0);
                wait_async0();   // LDS writes from the async pipe are visible
            }
            __syncthreads();
            for (int kk = 0; kk < KS; kk += 4) {
                int kb = kk + hi * 2;
                v2f aS, aV, bV;
                // A: 16x4, lanes 0-15 hold K={kb,kb+1}, lanes 16-31 K={kb+2,kb+3}
                aS.x = sS[mi * 16 + l16][kb];  aS.y = sS[mi * 16 + l16][kb + 1];
                aV.x = sV[mi * 16 + l16][kb];  aV.y = sV[mi * 16 + l16][kb + 1];
                // B: 4x16 (= V^T slice), mirrored layout
                bV.x = sV[ni * 16 + l16][kb];  bV.y = sV[ni * 16 + l16][kb + 1];
                acc1 = __builtin_amdgcn_wmma_f32_16x16x4_f32(
                    false, aS, false, bV, (short)0, acc1, false, false);
                acc2 = __builtin_amdgcn_wmma_f32_16x16x4_f32(
                    false, aV, false, bV, (short)0, acc2, false, false);
            }
        }
        __syncthreads();
        // C/D layout: VGPR r -> M = r (lanes 0-15) / r+8 (lanes 16-31), N = lane%16
        for (int r = 0; r < 8; ++r) {
            int m = mi * 16 + r + hi * 8;
            int n = ni * 16 + l16;
            simsT[m][n] = acc1[r];
            gram[m][n]  = acc2[r];
        }
        __syncthreads();

        // ---- sequential within-chunk correction (one wave; lane owns 2 classes)
        if (tid < 32) {
            for (int j = 0; j < CH; ++j) {
                int   lj  = lab[j];
                float vn2 = gram[j][j];
                float nv  = fmaxf(sqrtf(vn2), EPSV);
                for (int h = 0; h < 2; ++h) {
                    int k  = lane + h * 32;
                    int ws = wasSeen[k];
                    int fi = firstIn[k];
                    // base term: chunk-start prototype, unless class was reset in-chunk
                    float dotv = (ws || fi >= j) ? simsT[k][j] : 0.f;
                    int i0 = ws ? 0 : fi;
                    for (int i = i0; i < j; ++i)
                        if (lab[i] == k) dotv += gram[i][j];
                    float nk = fmaxf(sqrtf(n2[k]), EPSV);
                    out[(size_t)(base + j) * KK + k] = dotv / (nk * nv);
                    if (k == lj) {  // state update for the labeled class
                        int seenNow = ws || (fi < 64);
                        if (seenNow) n2[k] = n2[k] + 2.f * dotv + vn2;
                        else         { n2[k] = vn2; firstIn[k] = j; }
                    }
                }
            }
        }
        __syncthreads();

        // ---- fold chunk tokens into prototype state S (replace on first-seen)
        for (int d = tid; d < DD; d += (int)blockDim.x) {
            for (int j = 0; j < CH; ++j) {
                int   l = lab[j];
                float v = inputs[(size_t)(base + j) * DD + d];
                if (!wasSeen[l] && firstIn[l] == j) S[(size_t)l * DD + d] = v;
                else                               S[(size_t)l * DD + d] += v;
            }
        }
        if (tid < KK && firstIn[tid] < 64) seenP[tid] = 1;
        __syncthreads();
    }
}

extern "C" void kernel_launch(void* const* d_in, const int* in_sizes, int n_in,
                              void* d_out, int out_size, void* d_ws, size_t ws_size,
                              hipStream_t stream) {
    const float* inputs     = (const float*)d_in[0];
    const float* class_avgs = (const float*)d_in[1];
    const int*   labels     = (const int*)d_in[2];
    float*       out        = (float*)d_out;
    float*       S          = (float*)d_ws;   // 64*2048 floats = 512 KB of scratch
    simrel_cdna5_kernel<<<dim3(1), dim3(512), 0, stream>>>(
        inputs, class_avgs, labels, out, S);
}